// DeepMultiOmicPathwayNet_39436389712058
// MI455X (gfx1250) — compile-verified
//
#include <hip/hip_runtime.h>
#include <hip/hip_bf16.h>
#include <math.h>

typedef __attribute__((ext_vector_type(2))) float v2f;
typedef __attribute__((ext_vector_type(8))) float v8f;

#define NN   1024
#define GG   20000
#define CC   3
#define PP   200
#define KK   64
#define KC   192      // KK*CC
#define NC   5000
#define NCK  15000    // NC*CC
#define HID  512
#define OUT  20
#define FEAT 13312    // PP*KK + HID

static __device__ __forceinline__ float sigmoidf_(float v) {
    return 1.0f / (1.0f + __expf(-v));
}

// ---------------------------------------------------------------------------
// Kernel 0: out[n, o] = b_out[o]  (atomic accumulation target)
// ---------------------------------------------------------------------------
__global__ __launch_bounds__(256)
void out_init_kernel(const float* __restrict__ b_out, float* __restrict__ out) {
    const int i = blockIdx.x * 256 + threadIdx.x;
    if (i < NN * OUT) out[i] = b_out[i % OUT];
}

// ---------------------------------------------------------------------------
// Kernel A: pathway branch, fully fused.
// grid = (P=200, N/16=64), block = 128 (4 waves).
// Per block: GEMM 16x64x192 (f32 WMMA) -> bias -> per-row L2 norm -> sigmoid,
// then the 16x64 activation tile is immediately contracted against its
// 64x20 slice of W_out (second WMMA pass, cols padded to 32) and accumulated
// into `out` with global_atomic_add_f32.
// ---------------------------------------------------------------------------
__global__ __launch_bounds__(128)
void pathway_kernel(const float* __restrict__ x,
                    const int*   __restrict__ pathway_ids,
                    const float* __restrict__ W_path,
                    const float* __restrict__ b_path,
                    const float* __restrict__ W_out,
                    float*       __restrict__ out) {
    const int p    = blockIdx.x;
    const int nb   = blockIdx.y;          // batch tile (16 rows)
    const int tid  = threadIdx.x;
    const int wave = tid >> 5;            // 0..3 -> output cols 16*wave..
    const int lane = tid & 31;

    __shared__ int   ids[KK];
    __shared__ float As[16][KC];          // gathered A tile (12 KB)
    __shared__ float Ts[16][KK];          // activation tile (4 KB)
    __shared__ float Ws[KK][32];          // W_out slice, zero-padded (8 KB)
    __shared__ float rnorm[16];

    if (tid < KK) ids[tid] = pathway_ids[p * KK + tid];
    __syncthreads();

    // Gather 16 rows x 64 genes x 3 channels into LDS (3 contiguous floats).
    for (int idx = tid; idx < 16 * KK; idx += 128) {
        const int m = idx >> 6;
        const int j = idx & 63;
        const float* src = x + ((size_t)(nb * 16 + m) * GG + ids[j]) * CC;
        As[m][j * 3 + 0] = src[0];
        As[m][j * 3 + 1] = src[1];
        As[m][j * 3 + 2] = src[2];
    }
    // Stage W_out slice for this pathway's 64 features (pad 20 -> 32 cols).
    for (int idx = tid; idx < KK * 32; idx += 128) {
        const int kk = idx >> 5;
        const int nn = idx & 31;
        Ws[kk][nn] = (nn < OUT) ? W_out[(size_t)(p * KK + kk) * OUT + nn] : 0.0f;
    }
    __syncthreads();

    // GEMM 1: 16x16 slice of the 16x64x192 pathway GEMM.
    const int    col0  = wave * 16;
    const int    row   = lane & 15;       // M index (A) / N index (B)
    const int    khalf = lane >> 4;       // 0: k,k+1   1: k+2,k+3
    const float* Wp    = W_path + (size_t)p * KC * KK;

    v8f acc = {};
    #pragma unroll 4
    for (int s = 0; s < KC / 4; ++s) {
        const int kb = 4 * s + 2 * khalf;
        v2f a; a.x = As[row][kb];
               a.y = As[row][kb + 1];
        v2f b; b.x = Wp[(size_t)kb * KK + col0 + row];
               b.y = Wp[(size_t)(kb + 1) * KK + col0 + row];
        acc = __builtin_amdgcn_wmma_f32_16x16x4_f32(false, a, false, b,
                                                    (short)0, acc, false, false);
    }

    // Spill C tile (+bias) to LDS: VGPR r -> M=r (lanes 0-15) / M=r+8 (16-31).
    const int mofs = (lane < 16) ? 0 : 8;
    #pragma unroll
    for (int r = 0; r < 8; ++r) {
        Ts[r + mofs][col0 + row] = acc[r] + b_path[p * KK + col0 + row];
    }
    __syncthreads();

    if (tid < 16) {
        float s = 0.0f;
        #pragma unroll 8
        for (int o = 0; o < KK; ++o) { const float v = Ts[tid][o]; s += v * v; }
        rnorm[tid] = rsqrtf(s);
    }
    __syncthreads();

    // Normalize + sigmoid in place.
    for (int idx = tid; idx < 16 * KK; idx += 128) {
        const int m = idx >> 6;
        const int o = idx & 63;
        Ts[m][o] = sigmoidf_(Ts[m][o] * rnorm[m]);
    }
    __syncthreads();

    // GEMM 2 (epilogue): Ts[16x64] @ Ws[64x32] -> atomic accumulate into out.
    if (wave < 2) {
        v8f acc2 = {};
        #pragma unroll
        for (int s = 0; s < KK / 4; ++s) {
            const int kb = 4 * s + 2 * khalf;
            v2f a; a.x = Ts[row][kb];        a.y = Ts[row][kb + 1];
            v2f b; b.x = Ws[kb][col0 + row]; b.y = Ws[kb + 1][col0 + row];
            acc2 = __builtin_amdgcn_wmma_f32_16x16x4_f32(false, a, false, b,
                                                         (short)0, acc2, false, false);
        }
        const int n = col0 + row;
        if (n < OUT) {
            #pragma unroll
            for (int r = 0; r < 8; ++r) {
                unsafeAtomicAdd(&out[(size_t)(nb * 16 + r + mofs) * OUT + n],
                                acc2[r]);
            }
        }
    }
}

// ---------------------------------------------------------------------------
// Kernel B: non-cancer branch, fully fused.
// grid = (1024/32, 512/64), block = 128 (4 waves). 32x64 output tile per
// block; K = 15000 staged in 64-chunks (A gathered through nc_ids on the fly,
// B coalesced). Epilogue: bias + sigmoid tile (reusing the A-staging LDS),
// then tile @ W_out[12800+n0 .., :20] -> atomic accumulate into out.
// ---------------------------------------------------------------------------
__global__ __launch_bounds__(128)
void nc_kernel(const float* __restrict__ x,
               const int*   __restrict__ nc_ids,
               const float* __restrict__ W_nc,
               const float* __restrict__ b_nc,
               const float* __restrict__ W_out,
               float*       __restrict__ out) {
    const int m0   = blockIdx.x * 32;     // batch rows
    const int n0   = blockIdx.y * 64;     // hidden cols
    const int tid  = threadIdx.x;
    const int wave = tid >> 5;
    const int lane = tid & 31;

    __shared__ float As[32][64];          // A chunk, later sigmoid tile (8 KB)
    __shared__ float Bs[64][64];          // W_nc chunk (16 KB)
    __shared__ float Ws[64][32];          // W_out slice, zero-padded (8 KB)

    const int col0  = wave * 16;
    const int row   = lane & 15;
    const int khalf = lane >> 4;

    v8f acc0 = {};
    v8f acc1 = {};

    const int nChunks = (NCK + 63) / 64;
    for (int ch = 0; ch < nChunks; ++ch) {
        const int kc = ch * 64;

        // Stage gathered A chunk: As[m][kk] = x[m0+m, nc_ids[(kc+kk)/3], (kc+kk)%3]
        for (int idx = tid; idx < 32 * 64; idx += 128) {
            const int m  = idx >> 6;
            const int kk = idx & 63;
            const int k  = kc + kk;
            float v = 0.0f;
            if (k < NCK) {
                const int j = k / 3;
                const int c = k - 3 * j;
                v = x[((size_t)(m0 + m) * GG + nc_ids[j]) * CC + c];
            }
            As[m][kk] = v;
        }
        // Stage B chunk (coalesced rows of W_nc).
        for (int idx = tid; idx < 64 * 64; idx += 128) {
            const int kk = idx >> 6;
            const int nn = idx & 63;
            const int k  = kc + kk;
            Bs[kk][nn] = (k < NCK) ? W_nc[(size_t)k * HID + n0 + nn] : 0.0f;
        }
        // Prefetch next B chunk into L2 (global_prefetch_b8).
        if (tid == 0 && kc + 64 < NCK) {
            __builtin_prefetch(&W_nc[(size_t)(kc + 64) * HID + n0], 0, 1);
        }
        __syncthreads();

        #pragma unroll
        for (int s = 0; s < 16; ++s) {
            const int kb = 4 * s + 2 * khalf;
            v2f b; b.x = Bs[kb][col0 + row];
                   b.y = Bs[kb + 1][col0 + row];
            v2f a0; a0.x = As[row][kb];      a0.y = As[row][kb + 1];
            v2f a1; a1.x = As[16 + row][kb]; a1.y = As[16 + row][kb + 1];
            acc0 = __builtin_amdgcn_wmma_f32_16x16x4_f32(false, a0, false, b,
                                                         (short)0, acc0, false, false);
            acc1 = __builtin_amdgcn_wmma_f32_16x16x4_f32(false, a1, false, b,
                                                         (short)0, acc1, false, false);
        }
        __syncthreads();
    }

    // Epilogue: bias + sigmoid into LDS tile (reuse As as 32x64 tile).
    {
        const float bias = b_nc[n0 + col0 + row];
        const int   mofs = (lane < 16) ? 0 : 8;
        #pragma unroll
        for (int r = 0; r < 8; ++r) {
            As[r + mofs][col0 + row]      = sigmoidf_(acc0[r] + bias);
            As[16 + r + mofs][col0 + row] = sigmoidf_(acc1[r] + bias);
        }
    }
    // Stage W_out slice for features 12800 + n0 + kk (pad 20 -> 32 cols).
    for (int idx = tid; idx < 64 * 32; idx += 128) {
        const int kk = idx >> 5;
        const int nn = idx & 31;
        Ws[kk][nn] = (nn < OUT)
                   ? W_out[(size_t)(PP * KK + n0 + kk) * OUT + nn] : 0.0f;
    }
    __syncthreads();

    // GEMM 2 (epilogue): Sig[32x64] @ Ws[64x32]; 4 waves cover 2x2 tiles.
    {
        const int mt    = wave >> 1;      // 0/1: M subtile
        const int col0b = (wave & 1) * 16;
        v8f acc2 = {};
        #pragma unroll
        for (int s = 0; s < 16; ++s) {
            const int kb = 4 * s + 2 * khalf;
            v2f a; a.x = As[mt * 16 + row][kb]; a.y = As[mt * 16 + row][kb + 1];
            v2f b; b.x = Ws[kb][col0b + row];   b.y = Ws[kb + 1][col0b + row];
            acc2 = __builtin_amdgcn_wmma_f32_16x16x4_f32(false, a, false, b,
                                                         (short)0, acc2, false, false);
        }
        const int n    = col0b + row;
        const int mofs = (lane < 16) ? 0 : 8;
        if (n < OUT) {
            #pragma unroll
            for (int r = 0; r < 8; ++r) {
                unsafeAtomicAdd(&out[(size_t)(m0 + mt * 16 + r + mofs) * OUT + n],
                                acc2[r]);
            }
        }
    }
}

// ---------------------------------------------------------------------------
extern "C" void kernel_launch(void* const* d_in, const int* in_sizes, int n_in,
                              void* d_out, int out_size, void* d_ws, size_t ws_size,
                              hipStream_t stream) {
    (void)in_sizes; (void)n_in; (void)out_size; (void)d_ws; (void)ws_size;

    const float* x           = (const float*)d_in[0];
    const int*   pathway_ids = (const int*)  d_in[1];
    const int*   nc_ids      = (const int*)  d_in[2];
    const float* W_path      = (const float*)d_in[3];
    const float* b_path      = (const float*)d_in[4];
    const float* W_nc        = (const float*)d_in[5];
    const float* b_nc        = (const float*)d_in[6];
    const float* W_out       = (const float*)d_in[7];
    const float* b_out       = (const float*)d_in[8];
    float*       out         = (float*)d_out;

    out_init_kernel<<<(NN * OUT + 255) / 256, 256, 0, stream>>>(b_out, out);
    pathway_kernel<<<dim3(PP, NN / 16), 128, 0, stream>>>(
        x, pathway_ids, W_path, b_path, W_out, out);
    nc_kernel<<<dim3(NN / 32, HID / 64), 128, 0, stream>>>(
        x, nc_ids, W_nc, b_nc, W_out, out);
}